// DTWLoss_60206851555473
// MI455X (gfx1250) — compile-verified
//
#include <hip/hip_runtime.h>
#include <math.h>

typedef float v2f __attribute__((ext_vector_type(2)));
typedef float v8f __attribute__((ext_vector_type(8)));

#define BIGF 1.0e10f
#define NN   512
#define DD   64
#define NB   32            // 512/16 tile blocks per dim
#define DG   (2 * NN - 1)  // 1023 anti-diagonals

// One workgroup per batch element. 512 threads = 16 waves (wave32).
// LDS: gbuf = two block-anti-diagonals of 16x16 Gram tiles (2*32*256 f32 = 64KB)
//      dp   = 3 rotating diagonal arrays, y2s = y-row norms.
__global__ __launch_bounds__(512) void sdtw_fused_kernel(
    const float* __restrict__ X, const float* __restrict__ Y,
    float* __restrict__ losses) {
  __shared__ float gbuf[2 * NB * 256];
  __shared__ float dp[3][NN];
  __shared__ float y2s[NN];

  const int b    = blockIdx.x;
  const int tid  = threadIdx.x;
  const int lane = tid & 31;
  const int waveU = __builtin_amdgcn_readfirstlane(tid >> 5);  // scalar wave id
  const float* xb = X + (size_t)b * NN * DD;
  const float* yb = Y + (size_t)b * NN * DD;

  // --- row norms: thread i owns x-row i (register) and y-row i (LDS) ---
  float x2 = 0.0f;
  {
    const float4* xr = (const float4*)(xb + tid * DD);
    const float4* yr = (const float4*)(yb + tid * DD);
    float ys = 0.0f;
#pragma unroll
    for (int k = 0; k < DD / 4; ++k) {
      float4 a = xr[k];
      float4 c = yr[k];
      x2 += a.x * a.x + a.y * a.y + a.z * a.z + a.w * a.w;
      ys += c.x * c.x + c.y * c.y + c.z * c.z + c.w * c.w;
    }
    y2s[tid] = ys;
  }
  dp[0][tid] = BIGF;
  dp[1][tid] = BIGF;
  dp[2][tid] = BIGF;
  __syncthreads();

  const int half = lane >> 4;   // which K-pair of the 16x4 fragment
  const int lrow = lane & 15;   // fragment row / column

  float cur = BIGF;
  for (int s = 0; s < 64; ++s) {
    // ---- produce Gram block-anti-diagonal s into gbuf[s&1] via WMMA ----
    if (s < 2 * NB - 1) {
      const int Imin = (s - (NB - 1) > 0) ? (s - (NB - 1)) : 0;
      const int Imax = (s < NB - 1) ? s : (NB - 1);
      float* dst = gbuf + (s & 1) * (NB * 256);
#pragma unroll 1
      for (int rep = 0; rep < 2; ++rep) {
        const int I = Imin + waveU + rep * 16;  // scalar per wave
        if (I <= Imax) {                        // uniform branch, EXEC stays full
          const int J = s - I;
          const float* xa = xb + (I * 16 + lrow) * DD;
          const float* ya = yb + (J * 16 + lrow) * DD;
          v8f acc = {0.f, 0.f, 0.f, 0.f, 0.f, 0.f, 0.f, 0.f};
#pragma unroll
          for (int kk = 0; kk < 16; ++kk) {
            const int k0 = kk * 4 + half * 2;
            v2f av = *(const v2f*)(xa + k0);  // A: 16x4 chunk of X rows
            v2f bv = *(const v2f*)(ya + k0);  // B: 4x16 chunk of Y^T
            acc = __builtin_amdgcn_wmma_f32_16x16x4_f32(
                false, av, false, bv, (short)0, acc, false, false);
          }
          float* dblk = dst + I * 256;
#pragma unroll
          for (int r = 0; r < 8; ++r)  // D layout: row = r + 8*half, col = lrow
            dblk[(r + 8 * half) * 16 + lrow] = acc[r];
        }
      }
    }
    __syncthreads();

    // ---- 16 DP anti-diagonal steps consuming gbuf ----
    const int dbase = s * 16;
#pragma unroll 1
    for (int dd = 0; dd < 16; ++dd) {
      const int d = dbase + dd;
      if (d >= DG) break;  // uniform across block
      float* curArr = dp[d % 3];
      const float* prev1 = dp[(d + 2) % 3];
      const float* prev2 = dp[(d + 1) % 3];
      const int j = d - tid;
      float val = BIGF;
      if (j >= 0 && j < NN) {
        const float a  = (tid > 0) ? prev2[tid - 1] : ((d == 0) ? 0.0f : BIGF);
        const float bb = (tid > 0) ? prev1[tid - 1] : BIGF;
        const float c  = prev1[tid];
        const float m  = fminf(a, fminf(bb, c));
        const float sm =
            m - __logf(__expf(m - a) + __expf(m - bb) + __expf(m - c));
        const int I = tid >> 4, J = j >> 4;
        const float G = gbuf[((I + J) & 1) * (NB * 256) + I * 256 +
                             (tid & 15) * 16 + (j & 15)];
        val = (x2 + y2s[j] - 2.0f * G) + sm;
      }
      cur = val;
      curArr[tid] = val;
      __syncthreads();  // one barrier per diagonal (3-buffer rotation)
    }
  }

  if (tid == NN - 1) losses[b] = cur;  // last diagonal, i = N-1
}

__global__ void sdtw_mean_kernel(const float* __restrict__ losses, int B,
                                 float* __restrict__ out) {
  if (threadIdx.x == 0) {
    float s = 0.0f;
    for (int i = 0; i < B; ++i) s += losses[i];
    out[0] = s / (float)B;
  }
}

extern "C" void kernel_launch(void* const* d_in, const int* in_sizes, int n_in,
                              void* d_out, int out_size, void* d_ws,
                              size_t ws_size, hipStream_t stream) {
  const float* X = (const float*)d_in[0];
  const float* Y = (const float*)d_in[1];
  const int B = in_sizes[0] / (NN * DD);  // 64
  float* losses = (float*)d_ws;           // B floats of scratch

  sdtw_fused_kernel<<<B, 512, 0, stream>>>(X, Y, losses);
  sdtw_mean_kernel<<<1, 64, 0, stream>>>(losses, B, (float*)d_out);
}